// MambaEncoderDecoder_55920474194507
// MI455X (gfx1250) — compile-verified
//
#include <hip/hip_runtime.h>
#include <hip/hip_bf16.h>
#include <stdint.h>

// ---------------- problem constants ----------------
#define B_SZ   512
#define L_SZ   256
#define D_MODEL 64
#define D_STATE 16
#define D_INNER 128
#define DT_RANK 4
#define N_IDX  2048
#define ROWS   (B_SZ * L_SZ)            // 131072
#define XPROJ_N (DT_RANK + 2 * D_STATE) // 36
#define SPITCH 40                       // padded LDS row pitch (floats)

typedef __attribute__((ext_vector_type(2))) float v2f;
typedef __attribute__((ext_vector_type(8))) float v8f;

// ---------------- WMMA helpers (V_WMMA_F32_16X16X4_F32) ----------------
static __device__ __forceinline__ v8f wmma4(v2f a, v2f b, v8f c) {
    return __builtin_amdgcn_wmma_f32_16x16x4_f32(
        /*neg_a=*/false, a, /*neg_b=*/false, b,
        /*c_mod=*/(short)0, c, /*reuse_a=*/false, /*reuse_b=*/false);
}

// A tile 16x4 (f32): M = lane&15; VGPR0/1 hold K={0,1} (low half) / K={2,3} (high half).
static __device__ __forceinline__ v2f load_a_tile(const float* A, int lda,
                                                  int r0, int k0, int lane) {
    int m  = lane & 15;
    int kb = (lane >> 4) << 1;
    const float* p = A + (size_t)(r0 + m) * lda + k0 + kb;
    v2f a; a.x = p[0]; a.y = p[1];
    return a;
}

// B tile 4x16 (f32): VGPR r, lane-half h -> K = r + 2h, N = lane&15.
static __device__ __forceinline__ v2f load_b_tile(const float* B, int ldb,
                                                  int k0, int c0, int lane) {
    int n  = lane & 15;
    int kb = (lane >> 4) << 1;
    v2f b;
    b.x = B[(size_t)(k0 + kb)     * ldb + c0 + n];
    b.y = B[(size_t)(k0 + kb + 1) * ldb + c0 + n];
    return b;
}

// D tile 16x16 f32: VGPR r, lane-half h -> M = r + 8h, N = lane&15.
static __device__ __forceinline__ void store_d_tile(float* D, int ldc,
                                                    int r0, int c0, int lane,
                                                    v8f c, float bias) {
    int m0 = r0 + ((lane >> 4) << 3);
    int n  = c0 + (lane & 15);
#pragma unroll
    for (int r = 0; r < 8; ++r)
        D[(size_t)(m0 + r) * ldc + n] = c[r] + bias;
}

// ---------------- kernel 0: fold exp_w/exp_b through in_proj_w ----------------
// Weff(4,256) = exp_w(4,64) @ in_proj_w(64,256); beff(256) = exp_b @ in_proj_w
__global__ void prep_weights_kernel(const float* __restrict__ exp_w,
                                    const float* __restrict__ exp_b,
                                    const float* __restrict__ in_proj_w,
                                    float* __restrict__ Weff,
                                    float* __restrict__ beff) {
    int n = threadIdx.x; // 0..255
    float bacc = 0.f, w0 = 0.f, w1 = 0.f, w2 = 0.f, w3 = 0.f;
    for (int k = 0; k < 64; ++k) {
        float ipw = in_proj_w[k * 256 + n];
        bacc += exp_b[k] * ipw;
        w0 += exp_w[0 * 64 + k] * ipw;
        w1 += exp_w[1 * 64 + k] * ipw;
        w2 += exp_w[2 * 64 + k] * ipw;
        w3 += exp_w[3 * 64 + k] * ipw;
    }
    beff[n] = bacc;
    Weff[0 * 256 + n] = w0; Weff[1 * 256 + n] = w1;
    Weff[2 * 256 + n] = w2; Weff[3 * 256 + n] = w3;
}

// ---------------- kernel 1: XZ(ROWS,256) = reads(ROWS,4) @ Weff + beff --------
__global__ void gemm_inproj_kernel(const float* __restrict__ reads,
                                   const float* __restrict__ Weff,
                                   const float* __restrict__ beff,
                                   float* __restrict__ XZ) {
    int lane = threadIdx.x & 31;
    int wave = blockIdx.x * (blockDim.x >> 5) + (threadIdx.x >> 5);
    int tileN = wave & 15;       // 256/16
    int tileM = wave >> 4;       // ROWS/16 = 8192
    int r0 = tileM * 16, c0 = tileN * 16;
    v2f a = load_a_tile(reads, 4, r0, 0, lane);
    v2f b = load_b_tile(Weff, 256, 0, c0, lane);
    v8f c = {};
    c = wmma4(a, b, c);
    float bias = beff[c0 + (lane & 15)];
    store_d_tile(XZ, 256, r0, c0, lane, c, bias);
}

// ---------------- kernel 2: depthwise causal conv(4) + silu -> xc -------------
__global__ void conv_silu_kernel(const float* __restrict__ XZ,
                                 const float* __restrict__ conv_w,
                                 const float* __restrict__ conv_b,
                                 float* __restrict__ xc) {
    size_t i = (size_t)blockIdx.x * blockDim.x + threadIdx.x; // B*L*128
    int d = (int)(i & 127);
    size_t bl = i >> 7;            // b*L + l
    int l = (int)(bl & (L_SZ - 1));
    size_t brow = bl - (size_t)l;  // b*L
    float acc = conv_b[d];
#pragma unroll
    for (int t = 0; t < 4; ++t) {
        int lt = l - 3 + t;
        if (lt >= 0)
            acc += XZ[(brow + lt) * 256 + d] * conv_w[d * 4 + t];
    }
    xc[bl * 128 + d] = acc / (1.f + __expf(-acc)); // silu
}

// ---------------- kernel 3: P(ROWS,36) = xc(ROWS,128) @ x_proj_w(128,36) ------
__global__ void gemm_xproj_kernel(const float* __restrict__ xc,
                                  const float* __restrict__ xpw,
                                  float* __restrict__ P) {
    int lane = threadIdx.x & 31;
    int wave = blockIdx.x * (blockDim.x >> 5) + (threadIdx.x >> 5);
    int tileN = wave % 3;       // 48 padded cols = 3 tiles (36 valid)
    int tileM = wave / 3;       // 8192 tiles
    int r0 = tileM * 16, c0 = tileN * 16;
    int n  = lane & 15;
    int kb = (lane >> 4) << 1;
    bool nok = (c0 + n) < XPROJ_N;
    v8f c = {};
    for (int k0 = 0; k0 < 128; k0 += 4) {
        v2f a = load_a_tile(xc, 128, r0, k0, lane);
        v2f b;
        b.x = nok ? xpw[(size_t)(k0 + kb)     * XPROJ_N + c0 + n] : 0.f;
        b.y = nok ? xpw[(size_t)(k0 + kb + 1) * XPROJ_N + c0 + n] : 0.f;
        c = wmma4(a, b, c);
    }
    if (nok) {
        int m0 = r0 + ((lane >> 4) << 3);
#pragma unroll
        for (int r = 0; r < 8; ++r)
            P[(size_t)(m0 + r) * XPROJ_N + c0 + n] = c[r];
    }
}

// ---------------- kernel 4: selective scan ------------------------------------
// block = one batch element b (128 threads = 4 waves, one thread per channel d).
// State h[16] lives in VGPRs. The shared per-timestep dt/B/C row (36 floats) is
// double-buffered in LDS via CDNA5 async global->LDS DMA (ASYNCcnt): the row for
// step l+1 streams in while step l computes; one barrier per step.
// Pooling commutes with out_proj, so only sum_l(G) and G[L-1] are kept.
__global__ void scan_kernel(const float* __restrict__ xc,
                            const float* __restrict__ XZ,
                            const float* __restrict__ P,
                            const float* __restrict__ dt_w,
                            const float* __restrict__ dt_b,
                            const float* __restrict__ A_log,
                            const float* __restrict__ D_skip,
                            float* __restrict__ Gmean,
                            float* __restrict__ Glast) {
    __shared__ float sP[2][SPITCH];
    int b = blockIdx.x;
    int d = threadIdx.x; // 0..127
    float An[16];
#pragma unroll
    for (int n = 0; n < 16; ++n) An[n] = -__expf(A_log[d * 16 + n]);
    float w0 = dt_w[0 * 128 + d], w1 = dt_w[1 * 128 + d];
    float w2 = dt_w[2 * 128 + d], w3 = dt_w[3 * 128 + d];
    float dtb = dt_b[d];
    float dsk = D_skip[d];
    float h[16];
#pragma unroll
    for (int n = 0; n < 16; ++n) h[n] = 0.f;
    float gsum = 0.f, glast = 0.f;

    size_t base  = (size_t)b * L_SZ;            // first row index (b*L)
    size_t prow0 = base * XPROJ_N;              // first P element (floats)
    uint64_t pbase = (uint64_t)(uintptr_t)P;    // SGPR base for GVS async loads
    // Low 32 bits of a generic pointer to LDS hold the LDS byte offset.
    uint32_t lds_base = (uint32_t)(uintptr_t)(&sP[0][0]);
    bool loader = (d < XPROJ_N);

    // prologue: DMA row 0 into buffer 0
    if (loader) {
        uint32_t ga = (uint32_t)((prow0 + d) * 4u);
        uint32_t la = lds_base + (uint32_t)d * 4u;
        asm volatile("global_load_async_to_lds_b32 %0, %1, %2"
                     :: "v"(la), "v"(ga), "s"(pbase) : "memory");
    }
    asm volatile("s_wait_asynccnt 0x0" ::: "memory");
    __syncthreads();

    for (int l = 0; l < L_SZ; ++l) {
        int cur = l & 1;
        // kick off DMA of next row into the other buffer (overlaps compute)
        if (loader && (l + 1 < L_SZ)) {
            uint32_t ga = (uint32_t)((prow0 + (size_t)(l + 1) * XPROJ_N + d) * 4u);
            uint32_t la = lds_base + (uint32_t)(((cur ^ 1) * SPITCH + d) * 4);
            asm volatile("global_load_async_to_lds_b32 %0, %1, %2"
                         :: "v"(la), "v"(ga), "s"(pbase) : "memory");
        }
        const float* sp = sP[cur];
        size_t row = base + l;
        float u = xc[row * 128 + d];
        float z = XZ[row * 256 + 128 + d];
        // delta = softplus(dt_row @ dt_w[:,d] + dt_b[d])
        float x = dtb + sp[0] * w0 + sp[1] * w1 + sp[2] * w2 + sp[3] * w3;
        float delta = (x > 20.f) ? x : log1pf(__expf(x));
        float du = delta * u;
        float y = 0.f;
#pragma unroll
        for (int n = 0; n < 16; ++n) {
            h[n] = __expf(delta * An[n]) * h[n] + du * sp[4 + n];
            y += h[n] * sp[20 + n];
        }
        y += u * dsk;
        float g = y * (z / (1.f + __expf(-z))); // y * silu(z)
        gsum += g;
        glast = g;
        // retire the in-flight DMA, publish next buffer to all waves
        asm volatile("s_wait_asynccnt 0x0" ::: "memory");
        __syncthreads();
    }
    Gmean[b * 128 + d] = gsum * (1.f / (float)L_SZ);
    Glast[b * 128 + d] = glast;
}

// ---------------- kernel 5: pooled(B,128) = [Gmean;Glast] @ out_proj_w --------
__global__ void pool_gemm_kernel(const float* __restrict__ Gmean,
                                 const float* __restrict__ Glast,
                                 const float* __restrict__ out_proj_w,
                                 float* __restrict__ pooled) {
    int lane = threadIdx.x & 31;
    int wave = blockIdx.x * (blockDim.x >> 5) + (threadIdx.x >> 5); // 0..255
    int half  = wave & 1;          // 0: mean-pool, 1: last-pool
    int tileN = (wave >> 1) & 3;   // 64/16
    int tileM = wave >> 3;         // 512/16
    const float* A = half ? Glast : Gmean;
    int r0 = tileM * 16, c0 = tileN * 16;
    v8f c = {};
    for (int k0 = 0; k0 < 128; k0 += 4) {
        v2f a = load_a_tile(A, 128, r0, k0, lane);
        v2f b = load_b_tile(out_proj_w, 64, k0, c0, lane);
        c = wmma4(a, b, c);
    }
    int m0 = r0 + ((lane >> 4) << 3);
    int n  = c0 + (lane & 15) + half * 64;
#pragma unroll
    for (int r = 0; r < 8; ++r)
        pooled[(size_t)(m0 + r) * 128 + n] = c[r];
}

// ---------------- kernel 6: keys(512,64) = pooled @ k_w + k_b -----------------
__global__ void keys_gemm_kernel(const float* __restrict__ pooled,
                                 const float* __restrict__ k_w,
                                 const float* __restrict__ k_b,
                                 float* __restrict__ keys) {
    int lane = threadIdx.x & 31;
    int wave = blockIdx.x * (blockDim.x >> 5) + (threadIdx.x >> 5); // 0..127
    int tileN = wave & 3;
    int tileM = wave >> 2; // 32
    int r0 = tileM * 16, c0 = tileN * 16;
    v8f c = {};
    for (int k0 = 0; k0 < 128; k0 += 4) {
        v2f a = load_a_tile(pooled, 128, r0, k0, lane);
        v2f b = load_b_tile(k_w, 64, k0, c0, lane);
        c = wmma4(a, b, c);
    }
    store_d_tile(keys, 64, r0, c0, lane, c, k_b[c0 + (lane & 15)]);
}

// ------- kernel 7: queries(2048,64) = pooled[idx] @ q_w + q_b (gathered A) ----
__global__ void queries_gemm_kernel(const float* __restrict__ pooled,
                                    const int* __restrict__ idx,
                                    const float* __restrict__ q_w,
                                    const float* __restrict__ q_b,
                                    float* __restrict__ queries) {
    int lane = threadIdx.x & 31;
    int wave = blockIdx.x * (blockDim.x >> 5) + (threadIdx.x >> 5); // 0..511
    int tileN = wave & 3;
    int tileM = wave >> 2; // 128
    int r0 = tileM * 16, c0 = tileN * 16;
    int m  = lane & 15;
    int kb = (lane >> 4) << 1;
    size_t grow = (size_t)idx[r0 + m] * 128; // gather row of pooled
    v8f c = {};
    for (int k0 = 0; k0 < 128; k0 += 4) {
        v2f a; a.x = pooled[grow + k0 + kb]; a.y = pooled[grow + k0 + kb + 1];
        v2f b = load_b_tile(q_w, 64, k0, c0, lane);
        c = wmma4(a, b, c);
    }
    store_d_tile(queries, 64, r0, c0, lane, c, q_b[c0 + (lane & 15)]);
}

// ---------------- kernel 8: out(2048,512) = queries @ keys^T ------------------
__global__ void final_gemm_kernel(const float* __restrict__ queries,
                                  const float* __restrict__ keys,
                                  float* __restrict__ out) {
    int lane = threadIdx.x & 31;
    int wave = blockIdx.x * (blockDim.x >> 5) + (threadIdx.x >> 5); // 0..4095
    int tileN = wave & 31;  // 512/16
    int tileM = wave >> 5;  // 2048/16
    int r0 = tileM * 16, c0 = tileN * 16;
    int n  = lane & 15;
    int kb = (lane >> 4) << 1;
    v8f c = {};
    for (int k0 = 0; k0 < 64; k0 += 4) {
        v2f a = load_a_tile(queries, 64, r0, k0, lane);
        v2f b; // B(k, n) = keys[c0+n][k]
        b.x = keys[(size_t)(c0 + n) * 64 + k0 + kb];
        b.y = keys[(size_t)(c0 + n) * 64 + k0 + kb + 1];
        c = wmma4(a, b, c);
    }
    store_d_tile(out, 512, r0, c0, lane, c, 0.f);
}

// ---------------- launch ------------------------------------------------------
extern "C" void kernel_launch(void* const* d_in, const int* in_sizes, int n_in,
                              void* d_out, int out_size, void* d_ws, size_t ws_size,
                              hipStream_t stream) {
    const float* reads     = (const float*)d_in[0];
    const int*   seq_idx   = (const int*)  d_in[1];
    const float* exp_w     = (const float*)d_in[2];
    const float* exp_b     = (const float*)d_in[3];
    const float* in_proj_w = (const float*)d_in[4];
    const float* conv_w    = (const float*)d_in[5];
    const float* conv_b    = (const float*)d_in[6];
    const float* x_proj_w  = (const float*)d_in[7];
    const float* dt_w      = (const float*)d_in[8];
    const float* dt_b      = (const float*)d_in[9];
    const float* A_log     = (const float*)d_in[10];
    const float* D_skip    = (const float*)d_in[11];
    const float* out_proj_w= (const float*)d_in[12];
    const float* q_w       = (const float*)d_in[13];
    const float* q_b       = (const float*)d_in[14];
    const float* k_w       = (const float*)d_in[15];
    const float* k_b       = (const float*)d_in[16];
    float* out = (float*)d_out;

    // workspace layout (floats)
    float* ws = (float*)d_ws;
    constexpr size_t off_Weff    = 0;
    constexpr size_t off_beff    = off_Weff + 4 * 256;
    constexpr size_t off_XZ      = 2048;
    constexpr size_t off_xc      = off_XZ   + (size_t)ROWS * 256;
    constexpr size_t off_P       = off_xc   + (size_t)ROWS * 128;
    constexpr size_t off_Gmean   = off_P    + (size_t)ROWS * XPROJ_N;
    constexpr size_t off_Glast   = off_Gmean + (size_t)B_SZ * 128;
    constexpr size_t off_pooled  = off_Glast + (size_t)B_SZ * 128;
    constexpr size_t off_keys    = off_pooled + (size_t)B_SZ * 128;
    constexpr size_t off_queries = off_keys   + (size_t)B_SZ * 64;

    float* Weff    = ws + off_Weff;
    float* beff    = ws + off_beff;
    float* XZ      = ws + off_XZ;
    float* xc      = ws + off_xc;
    float* P       = ws + off_P;
    float* Gmean   = ws + off_Gmean;
    float* Glast   = ws + off_Glast;
    float* pooled  = ws + off_pooled;
    float* keys    = ws + off_keys;
    float* queries = ws + off_queries;

    prep_weights_kernel<<<1, 256, 0, stream>>>(exp_w, exp_b, in_proj_w, Weff, beff);

    // 8192 Mtiles x 16 Ntiles = 131072 waves; 8 waves/block
    gemm_inproj_kernel<<<16384, 256, 0, stream>>>(reads, Weff, beff, XZ);

    // B*L*128 elements
    conv_silu_kernel<<<65536, 256, 0, stream>>>(XZ, conv_w, conv_b, xc);

    // 8192 Mtiles x 3 Ntiles = 24576 waves
    gemm_xproj_kernel<<<3072, 256, 0, stream>>>(xc, x_proj_w, P);

    // one block per batch element, 128 threads (4 waves)
    scan_kernel<<<B_SZ, 128, 0, stream>>>(xc, XZ, P, dt_w, dt_b, A_log, D_skip,
                                          Gmean, Glast);

    // 32 Mtiles x 4 Ntiles x 2 halves = 256 waves
    pool_gemm_kernel<<<32, 256, 0, stream>>>(Gmean, Glast, out_proj_w, pooled);

    // 32 x 4 = 128 waves
    keys_gemm_kernel<<<16, 256, 0, stream>>>(pooled, k_w, k_b, keys);

    // 128 x 4 = 512 waves
    queries_gemm_kernel<<<64, 256, 0, stream>>>(pooled, seq_idx, q_w, q_b, queries);

    // 128 x 32 = 4096 waves
    final_gemm_kernel<<<512, 256, 0, stream>>>(queries, keys, out);
}